// CDLoss_17540646436952
// MI455X (gfx1250) — compile-verified
//
#include <hip/hip_runtime.h>
#include <hip/hip_bf16.h>

typedef float v2f __attribute__((ext_vector_type(2)));
typedef float v8f __attribute__((ext_vector_type(8)));

#define TPB            256   // 8 waves (wave32)
#define ROWS_PER_BLOCK 128   // 8 waves * 16 rows each
#define CHUNK          2048  // Q points staged in LDS per pass (32 KB)

// For each point p in P, find min over q in Q of max(|p-q|^2, 0), and emit the
// per-block sum of those mins into partials[blockIdx.x].
//
// d(p,q) = |p|^2 + |q|^2 - 2 p.q  computed via V_WMMA_F32_16X16X4_F32 with
//   A row = (x, y, z, 1)                       (16x4, K=4)
//   B col = (-2x', -2y', -2z', |q|^2)          (4x16)
// so D = |q|^2 - 2 p.q ; |p|^2 is a row constant added after the min
// (min and the monotone clamp both commute with min).
//
// B operand is staged in LDS pre-split by K-pair so each lane's per-tile load
// is a single ds_load_b64 with no selects:
//   sB[i]         = (-2x', -2y')   -> K0,K1 (consumed by lanes 0-15)
//   sB[CHUNK + i] = (-2z', |q|^2)  -> K2,K3 (consumed by lanes 16-31)
__global__ __launch_bounds__(TPB)
void chamfer_min_sum_kernel(const float* __restrict__ P,   // [B, N, 3]
                            const float* __restrict__ Q,   // [B, M, 3]
                            float* __restrict__ partials,  // [gridDim.x]
                            int N, int M)
{
    __shared__ float2 sB[2 * CHUNK];      // 32 KB
    __shared__ float  sWave[TPB / 32];

    const int blocksPerBatch = N / ROWS_PER_BLOCK;
    const int b    = blockIdx.x / blocksPerBatch;
    const int row0 = (blockIdx.x % blocksPerBatch) * ROWS_PER_BLOCK;

    const float* Pb = P + (size_t)b * N * 3;
    const float* Qb = Q + (size_t)b * M * 3;

    const int wave = threadIdx.x >> 5;
    const int lane = threadIdx.x & 31;
    const int half = lane >> 4;   // 0: lanes 0-15, 1: lanes 16-31
    const int l16  = lane & 15;

    // Each wave owns 16 rows of P; lane l16 (in both halves) loads point row0+wave*16+l16.
    const int   myRow = row0 + wave * 16 + l16;
    const float px = Pb[3 * (size_t)myRow + 0];
    const float py = Pb[3 * (size_t)myRow + 1];
    const float pz = Pb[3 * (size_t)myRow + 2];
    const float psq = px * px + py * py + pz * pz;

    // A-matrix 16x4 f32 layout: lanes 0-15 hold K=0 (VGPR0), K=1 (VGPR1);
    // lanes 16-31 hold K=2 (VGPR0), K=3 (VGPR1). Row M = l16.
    v2f A;
    A.x = half ? pz   : px;   // K=2 : K=0
    A.y = half ? 1.0f : py;   // K=3 : K=1

    // Loop-invariant per-lane B base: half selects which K-pair array we read.
    const float2* Bbase = sB + (half ? CHUNK : 0) + l16;

    // Running per-lane min of (|q|^2 - 2 p.q) in C/D layout:
    // VGPR g, lanes 0-15 -> row g ; lanes 16-31 -> row g+8.
    v8f mn;
#pragma unroll
    for (int g = 0; g < 8; ++g) mn[g] = 3.402823466e38f;

    for (int c0 = 0; c0 < M; c0 += CHUNK) {
        __syncthreads();
        // Stage CHUNK points of Q into LDS, pre-scaled and pre-split by K-pair.
        for (int i = threadIdx.x; i < CHUNK; i += TPB) {
            const float* q = Qb + 3 * (size_t)(c0 + i);
            float qx = q[0], qy = q[1], qz = q[2];
            sB[i]         = make_float2(-2.0f * qx, -2.0f * qy);
            sB[CHUNK + i] = make_float2(-2.0f * qz, qx * qx + qy * qy + qz * qz);
        }
        __syncthreads();

#pragma unroll 8
        for (int t = 0; t < CHUNK / 16; ++t) {
            v2f Bm = *(const v2f*)(Bbase + t * 16);   // one ds_load_b64, imm offset
            v8f c = {};
            v8f d = __builtin_amdgcn_wmma_f32_16x16x4_f32(
                false, A, false, Bm, (short)0, c, false, false);
#pragma unroll
            for (int g = 0; g < 8; ++g) mn[g] = fminf(mn[g], d[g]);
        }
    }

    // Finalize rows: add |p|^2 (row constant), clamp, min across the 16 lanes
    // of each half (one half covers rows 0-7, the other rows 8-15), then sum.
    float halfSum = 0.0f;
#pragma unroll
    for (int g = 0; g < 8; ++g) {
        // Row M = g + 8*half; its |p|^2 lives in lane l16 == M (either half).
        float sq1 = __shfl(psq, g + 8 * half, 32);
        float v = fmaxf(mn[g] + sq1, 0.0f);
        v = fminf(v, __shfl_xor(v, 1, 32));
        v = fminf(v, __shfl_xor(v, 2, 32));
        v = fminf(v, __shfl_xor(v, 4, 32));
        v = fminf(v, __shfl_xor(v, 8, 32));
        halfSum += v;   // identical in all 16 lanes of this half
    }
    float waveSum = halfSum + __shfl_xor(halfSum, 16, 32);  // sum of 16 row-mins

    if (lane == 0) sWave[wave] = waveSum;
    __syncthreads();
    if (threadIdx.x == 0) {
        float s = 0.0f;
#pragma unroll
        for (int w = 0; w < TPB / 32; ++w) s += sWave[w];
        partials[blockIdx.x] = s;
    }
}

// Deterministic fixed-order final reduction: out = mean(dist1) + mean(dist2).
__global__ void chamfer_finalize_kernel(const float* __restrict__ partials,
                                        float* __restrict__ out,
                                        int n1, int n2, float inv1, float inv2)
{
    if (threadIdx.x == 0 && blockIdx.x == 0) {
        float s1 = 0.0f, s2 = 0.0f;
        for (int i = 0; i < n1; ++i) s1 += partials[i];
        for (int i = 0; i < n2; ++i) s2 += partials[n1 + i];
        out[0] = s1 * inv1 + s2 * inv2;
    }
}

extern "C" void kernel_launch(void* const* d_in, const int* in_sizes, int n_in,
                              void* d_out, int out_size, void* d_ws, size_t ws_size,
                              hipStream_t stream)
{
    const float* pcs1 = (const float*)d_in[0];   // [8, 8192, 3] f32
    const float* pcs2 = (const float*)d_in[1];   // [8, 8192, 3] f32
    float* out      = (float*)d_out;             // scalar
    float* partials = (float*)d_ws;              // 2 * 512 floats = 4 KB

    const int B = 8;
    const int N = in_sizes[0] / (B * 3);         // 8192
    const int M = in_sizes[1] / (B * 3);         // 8192

    const int blocks1 = B * (N / ROWS_PER_BLOCK);  // 512
    const int blocks2 = B * (M / ROWS_PER_BLOCK);  // 512

    // Pass 1: for each p in pcs1, min over pcs2.
    chamfer_min_sum_kernel<<<blocks1, TPB, 0, stream>>>(pcs1, pcs2, partials, N, M);
    // Pass 2: roles swapped.
    chamfer_min_sum_kernel<<<blocks2, TPB, 0, stream>>>(pcs2, pcs1, partials + blocks1, M, N);
    // Deterministic finalize.
    chamfer_finalize_kernel<<<1, 64, 0, stream>>>(partials, out, blocks1, blocks2,
                                                  1.0f / (float)(B * N),
                                                  1.0f / (float)(B * M));
}